// AnchorSelector_16733192585220
// MI455X (gfx1250) — compile-verified
//
#include <hip/hip_runtime.h>
#include <hip/hip_bf16.h>
#include <stdint.h>

#define B_    8
#define N_    20000
#define D_    256
#define C_    91
#define G_    4
#define K_    300
#define CAP_  2048
#define P1_BLOCK 128
#define P1_RPW   32   // rows per wave (wave32)

#if defined(__has_builtin)
#if __has_builtin(__builtin_amdgcn_tensor_load_to_lds)
#define HAVE_TDM 1
#endif
#endif
#ifndef HAVE_TDM
#define HAVE_TDM 0
#endif

typedef unsigned int u32x4 __attribute__((ext_vector_type(4)));
typedef int          i32x4 __attribute__((ext_vector_type(4)));
typedef int          i32x8 __attribute__((ext_vector_type(8)));

// order-preserving float->u32 map: descending float == descending uint
__device__ __forceinline__ unsigned mono_key(float f) {
  unsigned u = __float_as_uint(f);
  return (u & 0x80000000u) ? ~u : (u | 0x80000000u);
}

// ---------------- Pass 1: per-row max of class_logits via TDM tile loads ---
__global__ __launch_bounds__(P1_BLOCK) void rowmax_kernel(
    const float* __restrict__ cls, unsigned* __restrict__ keys) {
  const int lane = threadIdx.x & 31;
  const int wave = threadIdx.x >> 5;
  const long long row0 = (long long)blockIdx.x * P1_BLOCK + (long long)wave * P1_RPW;
#if HAVE_TDM
  __shared__ __align__(16) float tile[P1_BLOCK * C_];
  // D# group0: count=1, lds_addr, global_addr[56:0] at bits[120:64], type=2
  unsigned lds_off = (unsigned)(uintptr_t)&tile[wave * P1_RPW * C_];
  unsigned long long ga = (unsigned long long)(uintptr_t)(cls + row0 * (long long)C_);
  u32x4 g0 = { 1u, lds_off, (unsigned)ga,
               (unsigned)((ga >> 32) & 0x1FFFFFFull) | 0x80000000u };
  // D# group1: data_size=4B(code 2); tensor=[91 x 32], stride0=91; tile=[91 x 32]
  i32x8 g1 = { 0x20000,          // wg_mask=0, data_size=2, no flags
               (C_ << 16),       // tensor_dim0[15:0] @ bits[63:48]
               (P1_RPW << 16),   // tensor_dim1[15:0] @ bits[95:80]
               (C_ << 16),       // tile_dim0 @ bits[127:112]
               P1_RPW,           // tile_dim1 @ bits[143:128]
               C_,               // tensor_dim0_stride[31:0]
               0, 0 };
  i32x4 g2 = { 0, 0, 0, 0 };
  i32x4 g3 = { 0, 0, 0, 0 };
#if defined(__clang_major__) && (__clang_major__ >= 23)
  i32x8 g4 = { 0, 0, 0, 0, 0, 0, 0, 0 };
  __builtin_amdgcn_tensor_load_to_lds(g0, g1, g2, g3, g4, 0);
#else
  __builtin_amdgcn_tensor_load_to_lds(g0, g1, g2, g3, 0);
#endif
  __builtin_amdgcn_s_wait_tensorcnt(0);
  asm volatile("" ::: "memory");  // LDS now holds the tile; block reordering
  const float* row = &tile[(wave * P1_RPW + lane) * C_];
#else
  const float* row = cls + (row0 + lane) * (long long)C_;
#endif
  float m = row[0];
#pragma unroll
  for (int c = 1; c < C_; ++c) m = fmaxf(m, row[c]);
  keys[row0 + lane] = mono_key(m);
}

// ---------------- Pass 2: exact deterministic top-K per batch ---------------
__global__ __launch_bounds__(1024) void topk_kernel(
    const unsigned* __restrict__ keys, int* __restrict__ topk) {
  const int b = blockIdx.x;
  const int tid = threadIdx.x;
  const unsigned* kb = keys + (size_t)b * N_;

  __shared__ unsigned hist[256];
  __shared__ unsigned scanbuf[1024];
  __shared__ unsigned long long cand[CAP_];
  __shared__ unsigned s_prefix, s_cgreater, s_cand;

  // byte-wise radix refinement until candidate set fits CAP_
  unsigned prefix = 0, cgreater = 0;
  int shift = 24;
  for (int level = 0; level < 4; ++level) {
    shift = 24 - 8 * level;
    if (tid < 256) hist[tid] = 0u;
    __syncthreads();
    for (int i = tid; i < N_; i += 1024) {
      unsigned k = kb[i];
      if (level == 0 || (k >> (shift + 8)) == (prefix >> (shift + 8)))
        atomicAdd(&hist[(k >> shift) & 0xFFu], 1u);
    }
    __syncthreads();
    if (tid == 0) {
      unsigned cum = cgreater, binc = 0;
      int v = 255;
      for (; v > 0; --v) {
        binc = hist[v];
        if (cum + binc >= (unsigned)K_) break;
        cum += binc;
      }
      if (v == 0) binc = hist[0];
      s_prefix   = prefix | ((unsigned)v << shift);
      s_cgreater = cum;
      s_cand     = cum + binc;
    }
    __syncthreads();
    prefix   = s_prefix;
    cgreater = s_cgreater;
    if (s_cand <= (unsigned)CAP_) break;  // uniform
    __syncthreads();
  }

  // deterministic scan-based compaction of candidates (top-bytes >= prefix)
  const unsigned pj = prefix >> shift;
  const int start = tid * 20;
  const int end = (start + 20 < N_) ? (start + 20) : N_;
  unsigned cnt = 0;
  for (int i = start; i < end; ++i) cnt += ((kb[i] >> shift) >= pj) ? 1u : 0u;
  scanbuf[tid] = cnt;
  __syncthreads();
  for (int off = 1; off < 1024; off <<= 1) {  // Hillis-Steele inclusive scan
    unsigned v = scanbuf[tid];
    unsigned add = (tid >= off) ? scanbuf[tid - off] : 0u;
    __syncthreads();
    scanbuf[tid] = v + add;
    __syncthreads();
  }
  unsigned total = scanbuf[1023];
  unsigned pos = scanbuf[tid] - cnt;
  for (int i = start; i < end; ++i) {
    unsigned k = kb[i];
    if ((k >> shift) >= pj) {
      if (pos < (unsigned)CAP_)
        cand[pos] = ((unsigned long long)k << 32) | (unsigned)(~(unsigned)i);
      ++pos;
    }
  }
  unsigned tot = (total < (unsigned)CAP_) ? total : (unsigned)CAP_;
  for (int i = tid; i < CAP_; i += 1024)
    if ((unsigned)i >= tot) cand[i] = 0ull;
  __syncthreads();

  // bitonic sort, descending by (value_key, ~index): exact jax top_k order
  for (unsigned sz = 2; sz <= (unsigned)CAP_; sz <<= 1) {
    for (unsigned j = sz >> 1; j > 0; j >>= 1) {
      __syncthreads();
      for (int i = tid; i < CAP_; i += 1024) {
        int ixj = i ^ (int)j;
        if (ixj > i) {
          unsigned long long a = cand[i];
          unsigned long long c = cand[ixj];
          bool desc = ((i & (int)sz) == 0);
          if ((a < c) == desc) { cand[i] = c; cand[ixj] = a; }
        }
      }
    }
  }
  __syncthreads();
  if (tid < K_)
    topk[b * K_ + tid] = (int)(~(unsigned)(cand[tid] & 0xFFFFFFFFull));
}

// ---------------- Pass 3: gather selected rows -----------------------------
__global__ __launch_bounds__(128) void gather_kernel(
    const float* __restrict__ mem, const float* __restrict__ cls,
    const float* __restrict__ geo, const int* __restrict__ topk,
    float* __restrict__ out) {
  const int t = blockIdx.x;   // 0..K_-1
  const int b = blockIdx.y;   // 0..B_-1
  const int tid = threadIdx.x;
  const int idx = topk[b * K_ + t];

  float* out_mem = out;
  float* out_cls = out + (size_t)B_ * K_ * D_;
  float* out_geo = out_cls + (size_t)B_ * K_ * C_;

  const float4* sm = (const float4*)(mem + ((size_t)b * N_ + (size_t)idx) * D_);
  float4* dm = (float4*)(out_mem + ((size_t)b * K_ + (size_t)t) * D_);
  if (tid < D_ / 4) dm[tid] = sm[tid];  // 64 float4 = 256 floats

  const float* sc = cls + ((size_t)b * N_ + (size_t)idx) * C_;
  float* dc = out_cls + ((size_t)b * K_ + (size_t)t) * C_;
  if (tid < C_) dc[tid] = sc[tid];

  const float* sg = geo + ((size_t)b * N_ + (size_t)idx) * G_;
  float* dg = out_geo + ((size_t)b * K_ + (size_t)t) * G_;
  if (tid < G_) dg[tid] = sg[tid];
}

// ---------------- Host-side launcher ---------------------------------------
extern "C" void kernel_launch(void* const* d_in, const int* in_sizes, int n_in,
                              void* d_out, int out_size, void* d_ws, size_t ws_size,
                              hipStream_t stream) {
  (void)in_sizes; (void)n_in; (void)out_size; (void)ws_size;
  const float* mem = (const float*)d_in[0];   // [8,20000,256]
  const float* cls = (const float*)d_in[1];   // [8,20000,91]
  const float* geo = (const float*)d_in[2];   // [8,20000,4]
  float* out = (float*)d_out;

  unsigned* keys = (unsigned*)d_ws;                                   // 160000 u32
  int* topk = (int*)((char*)d_ws + (size_t)B_ * N_ * sizeof(unsigned)); // 2400 i32

  const int p1_grid = (B_ * N_) / P1_BLOCK;  // 160000/128 = 1250 exactly
  rowmax_kernel<<<p1_grid, P1_BLOCK, 0, stream>>>(cls, keys);
  topk_kernel<<<B_, 1024, 0, stream>>>(keys, topk);
  gather_kernel<<<dim3(K_, B_), 128, 0, stream>>>(mem, cls, geo, topk, out);
}